// ROIAligner_16312285790451
// MI455X (gfx1250) — compile-verified
//
#include <hip/hip_runtime.h>
#include <cstdint>

// ROI-align, MI455X (gfx1250, wave32).
// Memory-bound kernel (~68 MB HBM traffic -> ~3us @ 23.3 TB/s). No matmul
// structure -> no WMMA; the CDNA5 path exercised is the async copy engine:
// GLOBAL_LOAD_ASYNC_TO_LDS_B128 double-buffering with S_WAIT_ASYNCCNT.

typedef float v4f __attribute__((ext_vector_type(4)));

#define FH 128
#define FW 128
#define FC 256
#define NB 8
#define NR 128
#define OH 7
#define OW 7
#define NPOS (NB * NR * OH * OW)   // 50176
#define WAVES_PER_BLOCK 8
#define BLOCK 256
#define NBLOCKS 784                 // 784*8 = 6272 waves; 50176/6272 = 8 iters/wave

struct PosCtx {
  const float* a[4];  // tl, tr, bl, br row base (channel 0)
  float lx, ly, m;
  float* op;          // output base (channel 0)
};

__device__ __forceinline__ PosCtx prepare(int p,
                                          const float* __restrict__ fmaps,
                                          const float* __restrict__ rois,
                                          float* __restrict__ out) {
  PosCtx c;
  const int b  = p / (NR * OH * OW);
  const int r  = p - b * (NR * OH * OW);
  const int n  = r / (OH * OW);
  const int ij = r - n * (OH * OW);
  const int i  = ij / OW;
  const int j  = ij - i * OW;

  const float* rp = rois + ((size_t)b * NR + n) * 4;
  const float rx1 = rp[0], ry1 = rp[1], rx2 = rp[2], ry2 = rp[3];

  // Replicate the reference's exact f32 op sequence.
  const float Hf = (float)FH, Wf = (float)FW;
  const float by1 = ry1 * Hf / (Hf - 1.0f);
  const float bx1 = rx1 * Wf / (Wf - 1.0f);
  const float by2 = ry2 * Hf / (Hf - 1.0f);
  const float bx2 = rx2 * Wf / (Wf - 1.0f);
  float in_y = by1 * (Hf - 1.0f) + (float)i * ((by2 - by1) * (Hf - 1.0f) / (float)(OH - 1));
  float in_x = bx1 * (Wf - 1.0f) + (float)j * ((bx2 - bx1) * (Wf - 1.0f) / (float)(OW - 1));
  const bool vy = (in_y >= 0.0f) && (in_y <= Hf - 1.0f);
  const bool vx = (in_x >= 0.0f) && (in_x <= Wf - 1.0f);
  in_y = fminf(fmaxf(in_y, 0.0f), Hf - 1.0f);
  in_x = fminf(fmaxf(in_x, 0.0f), Wf - 1.0f);
  const float y0 = floorf(in_y);
  const float x0 = floorf(in_x);
  c.ly = in_y - y0;
  c.lx = in_x - x0;
  const int y0i = (int)y0, x0i = (int)x0;
  const int y1i = min(y0i + 1, FH - 1);
  const int x1i = min(x0i + 1, FW - 1);

  const float* fb = fmaps + (size_t)b * FH * FW * FC;
  c.a[0] = fb + ((size_t)y0i * FW + x0i) * FC;
  c.a[1] = fb + ((size_t)y0i * FW + x1i) * FC;
  c.a[2] = fb + ((size_t)y1i * FW + x0i) * FC;
  c.a[3] = fb + ((size_t)y1i * FW + x1i) * FC;
  c.m  = (vy && vx) ? 1.0f : 0.0f;
  c.op = out + ((size_t)(b * NR + n) * (OH * OW) + ij) * FC;
  return c;
}

// One corner row: lane loads 32B (its 8 channels) via two async B128 ops.
// IOFFSET is added to BOTH the global and the LDS address (ISA 08_async §4.4),
// so a single (lds, global) address pair covers both 16B halves.
__device__ __forceinline__ void issue_corner(uint32_t lofs, const float* ga) {
  asm volatile("global_load_async_to_lds_b128 %0, %1, off"
               :: "v"(lofs), "v"(ga) : "memory");
  asm volatile("global_load_async_to_lds_b128 %0, %1, off offset:16"
               :: "v"(lofs), "v"(ga) : "memory");
}

__device__ __forceinline__ void issue_pos(const PosCtx& c, v4f* buf, int lane) {
#pragma unroll
  for (int k = 0; k < 4; ++k) {
    // Flat shared pointer's low 32 bits == wave-relative LDS byte address.
    const uint32_t lofs = (uint32_t)(uintptr_t)&buf[k * 64 + lane * 2];
    issue_corner(lofs, c.a[k] + lane * 8);
  }
}

__device__ __forceinline__ void blend_store(const PosCtx& c, const v4f* buf, int lane) {
  const float lx = c.lx, ly = c.ly, m = c.m;
#pragma unroll
  for (int k = 0; k < 2; ++k) {
    const v4f tl = buf[0 * 64 + lane * 2 + k];
    const v4f tr = buf[1 * 64 + lane * 2 + k];
    const v4f bl = buf[2 * 64 + lane * 2 + k];
    const v4f br = buf[3 * 64 + lane * 2 + k];
    const v4f top = tl + (tr - tl) * lx;
    const v4f bot = bl + (br - bl) * lx;
    const v4f res = (top + (bot - top) * ly) * m;
    // Output is write-once, never re-read: bypass L2 residency (NT store).
    __builtin_nontemporal_store(res, (v4f*)(c.op + lane * 8 + k * 4));
  }
}

__global__ __launch_bounds__(BLOCK)
void roi_align_async(const float* __restrict__ fmaps,
                     const float* __restrict__ rois,
                     float* __restrict__ out,
                     int npos, int stride) {
  // Wave-private double buffers: 4 corners x 1KB, x2. No barriers needed.
  __shared__ v4f stage[WAVES_PER_BLOCK][2][4 * 64];  // 64 KB / block

  const int lane = (int)(threadIdx.x & 31u);
  const int wave = __builtin_amdgcn_readfirstlane((int)(threadIdx.x >> 5));

  int pos = (int)blockIdx.x * WAVES_PER_BLOCK + wave;
  if (pos >= npos) return;

  v4f* const buf0 = &stage[wave][0][0];
  v4f* const buf1 = &stage[wave][1][0];

  PosCtx cur = prepare(pos, fmaps, rois, out);
  issue_pos(cur, buf0, lane);

  int dbl = 0;
  for (;;) {
    const int pnext = pos + stride;     // wave-uniform control flow
    const bool more = pnext < npos;
    PosCtx nxt = cur;
    if (more) {
      nxt = prepare(pnext, fmaps, rois, out);
      issue_pos(nxt, dbl ? buf0 : buf1, lane);
      // 8 newer loads outstanding; wait for the 8 older (current buffer).
      asm volatile("s_wait_asynccnt 0x8" ::: "memory");
    } else {
      asm volatile("s_wait_asynccnt 0x0" ::: "memory");
    }
    blend_store(cur, dbl ? buf1 : buf0, lane);
    if (!more) break;
    cur = nxt;
    dbl ^= 1;
    pos = pnext;
  }
}

extern "C" void kernel_launch(void* const* d_in, const int* in_sizes, int n_in,
                              void* d_out, int out_size, void* d_ws, size_t ws_size,
                              hipStream_t stream) {
  const float* fmaps = (const float*)d_in[0];  // [8,128,128,256] f32
  const float* rois  = (const float*)d_in[1];  // [8,128,4] f32
  float* out = (float*)d_out;                  // [8,128,7,7,256] f32
  (void)in_sizes; (void)n_in; (void)out_size; (void)d_ws; (void)ws_size;

  const int stride = NBLOCKS * WAVES_PER_BLOCK;  // 6272 -> exactly 8 iters/wave
  roi_align_async<<<NBLOCKS, BLOCK, 0, stream>>>(fmaps, rois, out, NPOS, stride);
}